// GanaGCN_27522150433354
// MI455X (gfx1250) — compile-verified
//
#include <hip/hip_runtime.h>
#include <hip/hip_bf16.h>
#include <math.h>

typedef float v2f __attribute__((ext_vector_type(2)));
typedef float v8f __attribute__((ext_vector_type(8)));

#define KDIM 64

// ---------------------------------------------------------------------------
// Degree / normalization: dinv[i] = rsqrt(1 + indegree(i))
// ---------------------------------------------------------------------------
__global__ void k_deg_init(float* deg, int n) {
    int i = blockIdx.x * blockDim.x + threadIdx.x;
    if (i < n) deg[i] = 1.0f;  // self-loop
}

__global__ void k_deg_count(const int* __restrict__ dst, float* deg, int e_cnt) {
    int stride = gridDim.x * blockDim.x;
    for (int e = blockIdx.x * blockDim.x + threadIdx.x; e < e_cnt; e += stride)
        atomicAdd(&deg[dst[e]], 1.0f);
}

__global__ void k_dinv(float* deg, int n) {
    int i = blockIdx.x * blockDim.x + threadIdx.x;
    if (i < n) deg[i] = __frsqrt_rn(deg[i]);  // deg >= 1 always
}

// ---------------------------------------------------------------------------
// Dense transform H = act(A) @ W via V_WMMA_F32_16X16X4_F32.
// A: [M x 64] row-major. W: [64 x NCOLS] row-major. H: [M x NCOLS].
// One wave per 16x16 output tile; wave w of the block owns column tile w,
// blockIdx.x owns the row tile. B fragment is preloaded into 32 VGPRs so the
// K-loop is a clean stream of global_load_b64(A) -> v_wmma.
// All B-edge handling is value-select (cndmask), never EXEC-divergent loads.
// ---------------------------------------------------------------------------
template <int NCOLS, bool RELU>
__global__ void k_gemm_wmma(const float* __restrict__ A,
                            const float* __restrict__ W,
                            float* __restrict__ H, int m) {
    const int lane  = threadIdx.x & 31;
    const int wave  = threadIdx.x >> 5;
    const int row0  = blockIdx.x * 16;
    const int col0  = wave * 16;
    if (row0 >= m) return;

    const int mrow  = lane & 15;      // M (for A) / N (for B) within tile
    const int khalf = lane >> 4;      // 0: K={0,1}, 1: K={2,3} within each K4

    int arow = row0 + mrow; if (arow >= m) arow = m - 1;  // value clamp only
    const float* arowp = A + (size_t)arow * KDIM + khalf * 2;

    const int  bcol = col0 + mrow;
    const bool bok  = (bcol < NCOLS);
    const int  bc   = bok ? bcol : (NCOLS - 1);  // clamped, always valid

    // --- preload B fragment: 16 x (2 floats) = full 64xK strip of this tile
    v2f bfrag[16];
    #pragma unroll
    for (int kt = 0; kt < 16; ++kt) {
        const int kbase = kt * 4 + khalf * 2;
        float bx = W[kbase * NCOLS + bc];
        float by = W[(kbase + 1) * NCOLS + bc];
        bfrag[kt].x = bok ? bx : 0.0f;   // v_cndmask, no branch
        bfrag[kt].y = bok ? by : 0.0f;
    }

    // --- K loop: 16 steps of 16x16x4
    v8f acc = {};
    #pragma unroll
    for (int kt = 0; kt < 16; ++kt) {
        v2f a;
        a.x = arowp[kt * 4];
        a.y = arowp[kt * 4 + 1];
        if (RELU) { a.x = fmaxf(a.x, 0.0f); a.y = fmaxf(a.y, 0.0f); }
        acc = __builtin_amdgcn_wmma_f32_16x16x4_f32(
            /*neg_a=*/false, a, /*neg_b=*/false, bfrag[kt],
            /*c_mod=*/(short)0, acc, /*reuse_a=*/false, /*reuse_b=*/false);
    }

    // --- store D: lane holds (N = col0 + lane&15, M = row0 + r + 8*(lane>>4))
    const int ccol  = col0 + (lane & 15);
    const int rbase = row0 + ((lane >> 4) << 3);
    if (ccol < NCOLS) {
        #pragma unroll
        for (int r = 0; r < 8; ++r) {
            int rr = rbase + r;
            if (rr < m) H[(size_t)rr * NCOLS + ccol] = acc[r];
        }
    }
}

// ---------------------------------------------------------------------------
// y = bias + dinv[i]^2 * h[i]   (self-loop term folded into init)
// ---------------------------------------------------------------------------
template <int NCOLS>
__global__ void k_agg_init(const float* __restrict__ h,
                           const float* __restrict__ dinv,
                           const float* __restrict__ bias,
                           float* __restrict__ y, int n) {
    int i = blockIdx.x * blockDim.x + threadIdx.x;
    if (i >= n * NCOLS) return;
    int node = i / NCOLS;
    int c    = i - node * NCOLS;
    float di = dinv[node];
    y[i] = bias[c] + di * di * h[i];
}

// ---------------------------------------------------------------------------
// Edge aggregation: one wave per edge, float2 per lane, L2-resident atomics.
// y[dst] += dinv[src]*dinv[dst] * h[src]
// ---------------------------------------------------------------------------
template <int NCOLS>
__global__ void k_agg_edges(const int* __restrict__ src,
                            const int* __restrict__ dst,
                            const float* __restrict__ dinv,
                            const float* __restrict__ h,
                            float* __restrict__ y, int e_cnt) {
    constexpr int pairs = NCOLS >> 1;
    const int lane   = threadIdx.x & 31;
    const int gwave  = (blockIdx.x * blockDim.x + threadIdx.x) >> 5;
    const int nwaves = (gridDim.x * blockDim.x) >> 5;
    for (int e = gwave; e < e_cnt; e += nwaves) {
        const int s = src[e];
        const int d = dst[e];
        const float norm = dinv[s] * dinv[d];
        if (lane < pairs) {
            const float2 hv = *(const float2*)(h + (size_t)s * NCOLS + (lane << 1));
            float* yp = y + (size_t)d * NCOLS + (lane << 1);
            atomicAdd(yp,     norm * hv.x);
            atomicAdd(yp + 1, norm * hv.y);
        }
    }
}

// ---------------------------------------------------------------------------
// Row-wise log_softmax over ncols=40: one wave per row, shuffle reductions.
// ---------------------------------------------------------------------------
template <int NCOLS>
__global__ void k_log_softmax(const float* __restrict__ y,
                              float* __restrict__ out, int n) {
    constexpr int pairs = NCOLS >> 1;  // 20
    const int lane = threadIdx.x & 31;
    const int row  = (blockIdx.x * blockDim.x + threadIdx.x) >> 5;
    if (row >= n) return;
    float vx = -INFINITY, vy = -INFINITY;
    if (lane < pairs) {
        const float2 v = *(const float2*)(y + (size_t)row * NCOLS + (lane << 1));
        vx = v.x; vy = v.y;
    }
    float mx = fmaxf(vx, vy);
    #pragma unroll
    for (int off = 16; off > 0; off >>= 1)
        mx = fmaxf(mx, __shfl_xor(mx, off, 32));
    float ex = (lane < pairs) ? (__expf(vx - mx) + __expf(vy - mx)) : 0.0f;
    #pragma unroll
    for (int off = 16; off > 0; off >>= 1)
        ex += __shfl_xor(ex, off, 32);
    const float lse = __logf(ex) + mx;
    if (lane < pairs) {
        float* o = out + (size_t)row * NCOLS + (lane << 1);
        o[0] = vx - lse;
        o[1] = vy - lse;
    }
}

// ---------------------------------------------------------------------------
extern "C" void kernel_launch(void* const* d_in, const int* in_sizes, int n_in,
                              void* d_out, int out_size, void* d_ws, size_t ws_size,
                              hipStream_t stream) {
    const float* x  = (const float*)d_in[0];
    const int*   ei = (const int*)  d_in[1];
    const float* W0 = (const float*)d_in[2];
    const float* b0 = (const float*)d_in[3];
    const float* W1 = (const float*)d_in[4];
    const float* b1 = (const float*)d_in[5];
    const float* W2 = (const float*)d_in[6];
    const float* b2 = (const float*)d_in[7];

    const int n = in_sizes[0] / KDIM;     // 50000
    const int e = in_sizes[1] / 2;        // 800000
    const int* src = ei;
    const int* dst = ei + e;

    float* dinv = (float*)d_ws;           // n floats
    float* hbuf = dinv + n;               // n*64 floats
    float* ybuf = hbuf + (size_t)n * 64;  // n*64 floats

    const int rowTiles = (n + 15) / 16;

    // --- normalization coefficients (once, reused by all 3 layers) ---
    k_deg_init<<<(n + 255) / 256, 256, 0, stream>>>(dinv, n);
    k_deg_count<<<1024, 256, 0, stream>>>(dst, dinv, e);
    k_dinv<<<(n + 255) / 256, 256, 0, stream>>>(dinv, n);

    // --- layer 0: h = x @ W0 ; y0 = Agg(h) + b0 ---
    k_gemm_wmma<64, false><<<rowTiles, 128, 0, stream>>>(x, W0, hbuf, n);
    k_agg_init<64><<<(n * 64 + 255) / 256, 256, 0, stream>>>(hbuf, dinv, b0, ybuf, n);
    k_agg_edges<64><<<2048, 256, 0, stream>>>(src, dst, dinv, hbuf, ybuf, e);

    // --- layer 1: h = relu(y0) @ W1 ; y1 = Agg(h) + b1 ---
    k_gemm_wmma<64, true><<<rowTiles, 128, 0, stream>>>(ybuf, W1, hbuf, n);
    k_agg_init<64><<<(n * 64 + 255) / 256, 256, 0, stream>>>(hbuf, dinv, b1, ybuf, n);
    k_agg_edges<64><<<2048, 256, 0, stream>>>(src, dst, dinv, hbuf, ybuf, e);

    // --- layer 2 (C=40): h = relu(y1) @ W2 ; y2 = Agg(h) + b2 ---
    k_gemm_wmma<40, true><<<rowTiles, 96, 0, stream>>>(ybuf, W2, hbuf, n);
    k_agg_init<40><<<(n * 40 + 255) / 256, 256, 0, stream>>>(hbuf, dinv, b2, ybuf, n);
    k_agg_edges<40><<<2048, 256, 0, stream>>>(src, dst, dinv, hbuf, ybuf, e);

    // --- log_softmax over pre-relu last-layer activations ---
    k_log_softmax<40><<<(n + 7) / 8, 256, 0, stream>>>(ybuf, (float*)d_out, n);
}